// CustomLoss_45603962748995
// MI455X (gfx1250) — compile-verified
//
#include <hip/hip_runtime.h>

// ---------------- problem constants (from reference) ----------------
#define BQ    256      // batch
#define DD    512      // feature dim
#define NX    50000    // pool points
#define KK    32       // kNN K
#define NPOOL 1024
#define TAU   0.1f
#define EPSI  0.1f
#define DELTA 1e-3f
#define SINK_ITERS 200
#define NS_ITERS   18
#define NCC   2048     // kNN distance chunk
#define SORTN 4096     // bitonic size for merge (2048+32 padded)
#define KSLICE 2500    // split-K slice for the X^T X gram (20 slices)

typedef __attribute__((ext_vector_type(2))) float v2f;
typedef __attribute__((ext_vector_type(8))) float v8f;

// ---------------- WMMA f32 16x16x4 helper (codegen-confirmed) ----------------
__device__ inline v8f wmma4(v2f a, v2f b, v8f c) {
#if __has_builtin(__builtin_amdgcn_wmma_f32_16x16x4_f32)
  // 8 args: (neg_a, A, neg_b, B, c_mod, C, reuse_a, reuse_b)
  return __builtin_amdgcn_wmma_f32_16x16x4_f32(false, a, false, b, (short)0, c,
                                               false, false);
#else
  for (int i = 0; i < 8; ++i) c[i] += a.x * b.x + a.y * b.y;  // host-pass stub
  return c;
#endif
}

// ============ 32x32 register-blocked NN GEMM: C = A[M,K] * B[K,N] ============
// one wave -> 32x32 tile (2x2 WMMA frags). M,N mult of 32, K mult of 4.
__global__ void gemm_nn32_kernel(float* __restrict__ C, const float* __restrict__ A,
                                 const float* __restrict__ Bm, int M, int Nn, int Kd) {
  int lane = threadIdx.x;
  int ntiles = Nn >> 5;
  int m0 = (blockIdx.x / ntiles) << 5;
  int n0 = (blockIdx.x % ntiles) << 5;
  int r  = lane & 15;
  int kk = (lane >> 4) << 1;   // 0 or 2
  v8f acc[2][2] = {};
#pragma unroll 2
  for (int k0 = 0; k0 < Kd; k0 += 4) {
    v2f a[2], b[2];
#pragma unroll
    for (int i = 0; i < 2; ++i) {
      a[i].x = A[(size_t)(m0 + 16 * i + r) * Kd + k0 + kk];
      a[i].y = A[(size_t)(m0 + 16 * i + r) * Kd + k0 + kk + 1];
      b[i].x = Bm[(size_t)(k0 + kk) * Nn + n0 + 16 * i + r];
      b[i].y = Bm[(size_t)(k0 + kk + 1) * Nn + n0 + 16 * i + r];
    }
#pragma unroll
    for (int i = 0; i < 2; ++i)
#pragma unroll
      for (int j = 0; j < 2; ++j) acc[i][j] = wmma4(a[i], b[j], acc[i][j]);
  }
  int rb = (lane >> 4) << 3;
#pragma unroll
  for (int i = 0; i < 2; ++i)
#pragma unroll
    for (int j = 0; j < 2; ++j)
#pragma unroll
      for (int q = 0; q < 8; ++q)
        C[(size_t)(m0 + 16 * i + rb + q) * Nn + n0 + 16 * j + (lane & 15)] =
            acc[i][j][q];
}

// ===== 64x64 register-blocked TN GEMM with split-K atomic accumulation =====
// C += A^T*B over a K slice. Asrc [K,M] rm (lda), Bsrc [K,N] rm (ldb).
// grid.x = (M/64)*(N/64)*numSlices; C must be pre-zeroed.
__global__ void gemm_tn64_kernel(float* __restrict__ C, const float* __restrict__ Asrc,
                                 const float* __restrict__ Bsrc, int M, int Nn, int Kd,
                                 int lda, int ldb, int kslice) {
  int lane = threadIdx.x;
  int ntiles = Nn >> 6;
  int tiles  = (M >> 6) * ntiles;
  int tile   = blockIdx.x % tiles;
  int slice  = blockIdx.x / tiles;
  int m0 = (tile / ntiles) << 6;
  int n0 = (tile % ntiles) << 6;
  int r  = lane & 15;
  int kk = (lane >> 4) << 1;
  int kbeg = slice * kslice;
  int kend = kbeg + kslice; if (kend > Kd) kend = Kd;
  v8f acc[4][4] = {};
  for (int k0 = kbeg; k0 < kend; k0 += 4) {
    if ((k0 & 255) == 0) {   // stream-ahead hint for the HBM pass over X
      __builtin_prefetch(Asrc + (size_t)(k0 + 256) * lda + m0 + r, 0, 1);
      __builtin_prefetch(Bsrc + (size_t)(k0 + 256) * ldb + n0 + r, 0, 1);
    }
    v2f a[4], b[4];
#pragma unroll
    for (int i = 0; i < 4; ++i) {
      a[i].x = Asrc[(size_t)(k0 + kk) * lda + m0 + 16 * i + r];
      a[i].y = Asrc[(size_t)(k0 + kk + 1) * lda + m0 + 16 * i + r];
      b[i].x = Bsrc[(size_t)(k0 + kk) * ldb + n0 + 16 * i + r];
      b[i].y = Bsrc[(size_t)(k0 + kk + 1) * ldb + n0 + 16 * i + r];
    }
#pragma unroll
    for (int i = 0; i < 4; ++i)
#pragma unroll
      for (int j = 0; j < 4; ++j) acc[i][j] = wmma4(a[i], b[j], acc[i][j]);
  }
  int rb = (lane >> 4) << 3;
#pragma unroll
  for (int i = 0; i < 4; ++i)
#pragma unroll
    for (int j = 0; j < 4; ++j)
#pragma unroll
      for (int q = 0; q < 8; ++q)
        atomicAdd(&C[(size_t)(m0 + 16 * i + rb + q) * Nn + n0 + 16 * j +
                     (lane & 15)],
                  acc[i][j][q]);
}

// ====== 32x32 blocked kNN distance chunk: d2 = |T|^2 - 2 T.X^T + |X|^2 ======
__global__ void gemm_d2_kernel(float* __restrict__ d2, const float* __restrict__ T,
                               const float* __restrict__ X, const float* __restrict__ Tn,
                               const float* __restrict__ Xn, int j0, int jcount) {
  int lane = threadIdx.x;
  int ntiles = NCC >> 5;   // 64
  int m0 = (blockIdx.x / ntiles) << 5;
  int n0 = (blockIdx.x % ntiles) << 5;
  int r  = lane & 15;
  int kk = (lane >> 4) << 1;
  int jc[2];
#pragma unroll
  for (int i = 0; i < 2; ++i) {
    jc[i] = j0 + n0 + 16 * i + r;
    if (jc[i] >= NX) jc[i] = NX - 1;         // clamp (masked in epilogue)
  }
  v8f acc[2][2] = {};
#pragma unroll 2
  for (int k0 = 0; k0 < DD; k0 += 4) {
    v2f a[2], b[2];
#pragma unroll
    for (int i = 0; i < 2; ++i) {
      a[i].x = T[(size_t)(m0 + 16 * i + r) * DD + k0 + kk];
      a[i].y = T[(size_t)(m0 + 16 * i + r) * DD + k0 + kk + 1];
      b[i].x = X[(size_t)jc[i] * DD + k0 + kk];       // B = X^T tile
      b[i].y = X[(size_t)jc[i] * DD + k0 + kk + 1];
    }
#pragma unroll
    for (int i = 0; i < 2; ++i)
#pragma unroll
      for (int j = 0; j < 2; ++j) acc[i][j] = wmma4(a[i], b[j], acc[i][j]);
  }
  int rb = (lane >> 4) << 3;
  const float INF = __builtin_inff();
#pragma unroll
  for (int i = 0; i < 2; ++i)
#pragma unroll
    for (int j = 0; j < 2; ++j) {
      int col = n0 + 16 * j + (lane & 15);
      bool valid = (col < jcount);
      float xn = valid ? Xn[j0 + col] : 0.f;
#pragma unroll
      for (int q = 0; q < 8; ++q) {
        int row = m0 + 16 * i + rb + q;
        float v = valid ? (Tn[row] - 2.f * acc[i][j][q] + xn) : INF;
        d2[(size_t)row * NCC + col] = v;
      }
    }
}

// ================= reductions / elementwise =================
__global__ void init_kernel(float* scal, float* topkd, int* topki) {
  int i = blockIdx.x * blockDim.x + threadIdx.x;
  if (i < 32) scal[i] = 0.f;
  if (i < BQ * KK) { topkd[i] = __builtin_inff(); topki[i] = 0; }
}

__global__ void zero_kernel(float* A, int n) {
  int i = blockIdx.x * blockDim.x + threadIdx.x;
  if (i < n) A[i] = 0.f;
}

__global__ void colmean_kernel(float* mu, const float* A, int rows, int cols) {
  int c = blockIdx.x * blockDim.x + threadIdx.x;
  if (c >= cols) return;
  float s = 0.f;
  for (int r = 0; r < rows; ++r) s += A[(size_t)r * cols + c];
  mu[c] = s / (float)rows;
}

__global__ void rownorm_kernel(float* out, const float* A, int cols) {
  int r = blockIdx.x, lane = threadIdx.x;  // 32 threads
  float s = 0.f;
  for (int c = lane; c < cols; c += 32) { float x = A[(size_t)r * cols + c]; s += x * x; }
  for (int o = 16; o > 0; o >>= 1) s += __shfl_xor(s, o, 32);
  if (lane == 0) out[r] = s;
}

__global__ void sumsq_kernel(float* acc, const float* A, int n) {
  int i = blockIdx.x * blockDim.x + threadIdx.x;
  float s = 0.f;
  for (; i < n; i += gridDim.x * blockDim.x) { float x = A[i]; s += x * x; }
  for (int o = 16; o > 0; o >>= 1) s += __shfl_xor(s, o, 32);
  if ((threadIdx.x & 31) == 0) atomicAdd(acc, s);
}

__global__ void cov_finalize_kernel(float* Sig, const float* G, const float* mu,
                                    float cnt) {
  int i = blockIdx.x * blockDim.x + threadIdx.x;
  if (i >= DD * DD) return;
  int r = i / DD, c = i % DD;
  Sig[i] = (G[i] - cnt * mu[r] * mu[c]) / (cnt - 1.f);
}

__global__ void trace_kernel(float* out, const float* A) {  // 1 block x 256
  __shared__ float red[256];
  float s = 0.f;
  for (int i = threadIdx.x; i < DD; i += 256) s += A[(size_t)i * DD + i];
  red[threadIdx.x] = s; __syncthreads();
  for (int o = 128; o > 0; o >>= 1) {
    if (threadIdx.x < o) red[threadIdx.x] += red[threadIdx.x + o];
    __syncthreads();
  }
  if (threadIdx.x == 0) out[0] = red[0];
}

__global__ void ridge_kernel(float* Sig, const float* tr, float delta) {
  int i = blockIdx.x * blockDim.x + threadIdx.x;
  if (i < DD) Sig[(size_t)i * DD + i] += tr[0] / (float)DD * delta;
}

__global__ void ns_init_kernel(float* Y, float* Z, const float* A, const float* ssq) {
  int i = blockIdx.x * blockDim.x + threadIdx.x;
  if (i >= DD * DD) return;
  float nf = sqrtf(ssq[0]);                   // Frobenius norm
  Y[i] = A[i] / nf;
  Z[i] = ((i / DD) == (i % DD)) ? 1.f : 0.f;
}

__global__ void ns_tk_kernel(float* P) {      // P <- 0.5*(3I - P)
  int i = blockIdx.x * blockDim.x + threadIdx.x;
  if (i >= DD * DD) return;
  float diag = ((i / DD) == (i % DD)) ? 3.f : 0.f;
  P[i] = 0.5f * (diag - P[i]);
}

__global__ void scale_pow_kernel(float* Out, const float* Y, const float* ssq) {
  int i = blockIdx.x * blockDim.x + threadIdx.x;
  if (i >= DD * DD) return;
  Out[i] = Y[i] * sqrtf(sqrtf(ssq[0]));       // sqrt(A) = Y_ns * ||A||_F^{1/2}
}

__global__ void symmetrize_kernel(float* Out, const float* A) {
  int i = blockIdx.x * blockDim.x + threadIdx.x;
  if (i >= DD * DD) return;
  int r = i / DD, c = i % DD;
  Out[i] = 0.5f * (A[i] + A[(size_t)c * DD + r]);
}

__global__ void meandiff_kernel(float* out, const float* mu_t, const float* mu_X) {
  __shared__ float red[256];
  float s = 0.f;
  for (int i = threadIdx.x; i < DD; i += 256) {
    float d = mu_t[i] - mu_X[i]; s += d * d;
  }
  red[threadIdx.x] = s; __syncthreads();
  for (int o = 128; o > 0; o >>= 1) {
    if (threadIdx.x < o) red[threadIdx.x] += red[threadIdx.x + o];
    __syncthreads();
  }
  if (threadIdx.x == 0) out[0] = red[0];
}

// ================= top-K merge (bitonic) =================
__device__ inline void bitonic_kv(float* key, int* val, int n, int tid, int nt) {
  for (int k = 2; k <= n; k <<= 1) {
    for (int j = k >> 1; j > 0; j >>= 1) {
      for (int i = tid; i < n; i += nt) {
        int ixj = i ^ j;
        if (ixj > i) {
          bool up = ((i & k) == 0);
          float ki = key[i], kj = key[ixj];
          bool sw = up ? (ki > kj) : (ki < kj);
          if (sw) {
            key[i] = kj; key[ixj] = ki;
            int t = val[i]; val[i] = val[ixj]; val[ixj] = t;
          }
        }
      }
      __syncthreads();
    }
  }
}

__device__ inline void bitonic_k(float* key, int n, int tid, int nt) {
  for (int k = 2; k <= n; k <<= 1) {
    for (int j = k >> 1; j > 0; j >>= 1) {
      for (int i = tid; i < n; i += nt) {
        int ixj = i ^ j;
        if (ixj > i) {
          bool up = ((i & k) == 0);
          float ki = key[i], kj = key[ixj];
          bool sw = up ? (ki > kj) : (ki < kj);
          if (sw) { key[i] = kj; key[ixj] = ki; }
        }
      }
      __syncthreads();
    }
  }
}

__global__ void topk_merge_kernel(float* topkd, int* topki, const float* d2, int j0) {
  __shared__ float key[SORTN];
  __shared__ int   val[SORTN];
  int b = blockIdx.x, tid = threadIdx.x;       // 256 threads
  for (int i = tid; i < SORTN; i += 256) {
    if (i < NCC)            { key[i] = d2[(size_t)b * NCC + i]; val[i] = j0 + i; }
    else if (i < NCC + KK)  { key[i] = topkd[b * KK + (i - NCC)];
                              val[i] = topki[b * KK + (i - NCC)]; }
    else                    { key[i] = __builtin_inff(); val[i] = -1; }
  }
  __syncthreads();
  bitonic_kv(key, val, SORTN, tid, 256);
  if (tid < KK) { topkd[b * KK + tid] = key[tid]; topki[b * KK + tid] = val[tid]; }
}

// ================= post weights (softmax over K) =================
__global__ void postw_kernel(float* post_w, const float* topkd) {
  int b = blockIdx.x, k = threadIdx.x;         // 32 threads
  float x = -topkd[b * KK + k] / TAU;
  float mx = x;
  for (int o = 16; o > 0; o >>= 1) mx = fmaxf(mx, __shfl_xor(mx, o, 32));
  float e = __expf(x - mx);
  float s = e;
  for (int o = 16; o > 0; o >>= 1) s += __shfl_xor(s, o, 32);
  float w = fmaxf(e / s, 1e-8f);
  float s2 = w;
  for (int o = 16; o > 0; o >>= 1) s2 += __shfl_xor(s2, o, 32);
  post_w[b * KK + k] = w / s2;
}

// ================= pairwise cost matrix over support (2K=64) =================
__global__ void costmat_kernel(float* Craw, const float* X, const float* Xn,
                               const int* topki, const int* pre_indices,
                               const int* q_indices) {
  __shared__ int   sup[64];
  __shared__ float sq[64];
  int b = blockIdx.x, tid = threadIdx.x;       // 256 threads
  if (tid < 64) {
    int idx = (tid < KK) ? pre_indices[q_indices[b] * KK + tid]
                         : topki[b * KK + tid - KK];
    sup[tid] = idx; sq[tid] = Xn[idx];
  }
  __syncthreads();
  for (int p = tid; p < 64 * 64; p += 256) {
    int i = p >> 6, j = p & 63;
    const float* xi = X + (size_t)sup[i] * DD;
    const float* xj = X + (size_t)sup[j] * DD;
    float dot = 0.f;
#pragma unroll 4
    for (int d = 0; d < DD; ++d) dot += xi[d] * xj[d];
    Craw[(size_t)b * 4096 + p] = fmaxf(sq[i] + sq[j] - 2.f * dot, 0.f);
  }
}

__global__ void median_kernel(float* med, const float* Craw) {
  __shared__ float key[4096];
  int b = blockIdx.x, tid = threadIdx.x;       // 256 threads
  for (int i = tid; i < 4096; i += 256) key[i] = Craw[(size_t)b * 4096 + i];
  __syncthreads();
  bitonic_k(key, 4096, tid, 256);
  if (tid == 0) med[b] = 0.5f * (key[2047] + key[2048]);
}

// ================= Sinkhorn (one block per batch item) =================
__global__ void sinkhorn_kernel(float* knn_acc, const float* Craw, const float* med,
                                const float* post_w, const float* pre_weights,
                                const int* q_indices) {
  __shared__ float Cn[4096];
  __shared__ float Km[4096];
  __shared__ float u[64], v[64], pv[64], qv[64], red[64];
  int b = blockIdx.x, t = threadIdx.x;          // 64 threads
  float m = med[b] + 1e-8f;
  for (int i = t; i < 4096; i += 64) {
    float c = Craw[(size_t)b * 4096 + i] / m;
    Cn[i] = c;
    Km[i] = __expf(-c / EPSI);
  }
  if (t < KK) {
    int qi = q_indices[b];
    red[t] = fmaxf(pre_weights[qi * KK + t], 1e-8f);
  }
  __syncthreads();
  float psum = 0.f;
  for (int i = 0; i < KK; ++i) psum += red[i];
  if (t < KK) { pv[t] = red[t] / psum; qv[t] = 0.f; }
  else        { pv[t] = 0.f; qv[t] = post_w[b * KK + (t - KK)]; }
  u[t] = 1.f; v[t] = 1.f;
  __syncthreads();
  for (int it = 0; it < SINK_ITERS; ++it) {
    float s = 0.f;
    for (int j = 0; j < 64; ++j) s += Km[t * 64 + j] * v[j];
    u[t] = pv[t] / (s + 1e-16f);
    __syncthreads();
    float s2 = 0.f;
    for (int i = 0; i < 64; ++i) s2 += Km[i * 64 + t] * u[i];
    v[t] = qv[t] / (s2 + 1e-16f);
    __syncthreads();
  }
  float part = 0.f;
  for (int j = 0; j < 64; ++j)
    part += u[t] * Km[t * 64 + j] * v[j] * Cn[t * 64 + j];
  red[t] = part; __syncthreads();
  if (t == 0) {
    float tot = 0.f;
    for (int i = 0; i < 64; ++i) tot += red[i];
    atomicAdd(knn_acc, tot / (float)BQ);
  }
}

// ================= final combine =================
__global__ void combine_kernel(float* out, const float* trSX, const float* trSt,
                               const float* trY2, const float* ssq2,
                               const float* tmean, const float* knn,
                               const float* wss) {
  if (threadIdx.x == 0 && blockIdx.x == 0) {
    float tr_sm = trY2[0] * sqrtf(sqrtf(ssq2[0]));
    float term_cov = trSX[0] + trSt[0] - 2.f * tr_sm;
    float loss_dist = tmean[0] + fmaxf(term_cov, 0.f);
    float loss_knn = knn[0];
    float loss_reg = 0.5f * wss[0];
    float total = loss_dist + loss_knn + 1e-4f * loss_reg;
    out[0] = total; out[1] = loss_dist; out[2] = loss_knn;
  }
}

// ================= host orchestration =================
extern "C" void kernel_launch(void* const* d_in, const int* in_sizes, int n_in,
                              void* d_out, int out_size, void* d_ws, size_t ws_size,
                              hipStream_t stream) {
  (void)in_sizes; (void)n_in; (void)out_size; (void)ws_size;
  const float* q_batch     = (const float*)d_in[0];
  const int*   q_indices   = (const int*)  d_in[1];
  const float* W           = (const float*)d_in[2];
  const float* X           = (const float*)d_in[3];
  const int*   pre_indices = (const int*)  d_in[4];
  const float* pre_weights = (const float*)d_in[5];
  float* out = (float*)d_out;

  // -------- workspace carve-out (floats) --------
  float* ws = (float*)d_ws;
  size_t off = 0;
  const size_t MM = (size_t)DD * DD;
  float* T    = ws + off; off += (size_t)BQ * DD;
  float* G    = ws + off; off += MM;     // gram scratch, later reused for M2
  float* SigX = ws + off; off += MM;
  float* SigT = ws + off; off += MM;
  float* Yb   = ws + off; off += MM;
  float* Zb   = ws + off; off += MM;
  float* Pb   = ws + off; off += MM;
  float* Ynb  = ws + off; off += MM;
  float* Znb  = ws + off; off += MM;
  float* st   = ws + off; off += MM;     // sqrt(Sigma_t)
  float* M1   = ws + off; off += MM;     // st * SigX
  float* Msym = ws + off; off += MM;
  float* d2c  = ws + off; off += (size_t)BQ * NCC;
  float* topkd = ws + off; off += (size_t)BQ * KK;
  int*   topki = (int*)(ws + off); off += (size_t)BQ * KK;
  float* Craw = ws + off; off += (size_t)BQ * 64 * 64;
  float* med  = ws + off; off += BQ;
  float* postw = ws + off; off += (size_t)BQ * KK;
  float* muX  = ws + off; off += DD;
  float* muT  = ws + off; off += DD;
  float* Tn   = ws + off; off += BQ;
  float* Xn   = ws + off; off += NX;
  float* scal = ws + off; off += 32;
  // scal: 0=trSX 1=trSt 2=ssq1 3=ssq2 4=trY2 5=tmean 6=knn 7=wss 8/9=pre-ridge tr

  const int EW = (DD * DD + 255) / 256;        // elementwise grid over 512x512
  const int G32 = (DD / 32) * (DD / 32);       // 256 blocks for 512x512 nn GEMM
  const int TN_TILES = (DD / 64) * (DD / 64);  // 64 tiles for 64x64 tn GEMM
  const int NSLICES = (NX + KSLICE - 1) / KSLICE;  // 20

  // -------- init accumulators / top-K --------
  init_kernel<<<(BQ * KK + 255) / 256, 256, 0, stream>>>(scal, topkd, topki);

  // -------- stats of X --------
  colmean_kernel<<<(DD + 255) / 256, 256, 0, stream>>>(muX, X, NX, DD);
  rownorm_kernel<<<NX, 32, 0, stream>>>(Xn, X, DD);

  // -------- T = q_batch @ W --------
  gemm_nn32_kernel<<<(BQ / 32) * (DD / 32), 32, 0, stream>>>(T, q_batch, W,
                                                             BQ, DD, DD);
  colmean_kernel<<<(DD + 255) / 256, 256, 0, stream>>>(muT, T, BQ, DD);
  rownorm_kernel<<<BQ, 32, 0, stream>>>(Tn, T, DD);

  // -------- Sigma_X = ridge(cov(X)) : split-K gram over N=50000 --------
  zero_kernel<<<EW, 256, 0, stream>>>(G, DD * DD);
  gemm_tn64_kernel<<<TN_TILES * NSLICES, 32, 0, stream>>>(G, X, X, DD, DD, NX,
                                                          DD, DD, KSLICE);
  cov_finalize_kernel<<<EW, 256, 0, stream>>>(SigX, G, muX, (float)NX);
  trace_kernel<<<1, 256, 0, stream>>>(scal + 8, SigX);
  ridge_kernel<<<(DD + 255) / 256, 256, 0, stream>>>(SigX, scal + 8, DELTA);
  trace_kernel<<<1, 256, 0, stream>>>(scal + 0, SigX);

  // -------- Sigma_t = ridge(cov(T)) --------
  zero_kernel<<<EW, 256, 0, stream>>>(G, DD * DD);
  gemm_tn64_kernel<<<TN_TILES, 32, 0, stream>>>(G, T, T, DD, DD, BQ, DD, DD, BQ);
  cov_finalize_kernel<<<EW, 256, 0, stream>>>(SigT, G, muT, (float)BQ);
  trace_kernel<<<1, 256, 0, stream>>>(scal + 9, SigT);
  ridge_kernel<<<(DD + 255) / 256, 256, 0, stream>>>(SigT, scal + 9, DELTA);
  trace_kernel<<<1, 256, 0, stream>>>(scal + 1, SigT);

  // -------- s_t = sqrt(Sigma_t) via coupled Newton-Schulz --------
  sumsq_kernel<<<256, 256, 0, stream>>>(scal + 2, SigT, DD * DD);
  ns_init_kernel<<<EW, 256, 0, stream>>>(Yb, Zb, SigT, scal + 2);
  {
    float *Y = Yb, *Z = Zb, *Yn2 = Ynb, *Zn2 = Znb;
    for (int it = 0; it < NS_ITERS; ++it) {
      gemm_nn32_kernel<<<G32, 32, 0, stream>>>(Pb, Z, Y, DD, DD, DD);
      ns_tk_kernel<<<EW, 256, 0, stream>>>(Pb);
      gemm_nn32_kernel<<<G32, 32, 0, stream>>>(Yn2, Y, Pb, DD, DD, DD);
      gemm_nn32_kernel<<<G32, 32, 0, stream>>>(Zn2, Pb, Z, DD, DD, DD);
      float* t1 = Y; Y = Yn2; Yn2 = t1;
      float* t2 = Z; Z = Zn2; Zn2 = t2;
    }
    scale_pow_kernel<<<EW, 256, 0, stream>>>(st, Y, scal + 2);
  }

  // -------- M = sym(s_t * Sigma_X * s_t), trace(sqrt(M)) --------
  gemm_nn32_kernel<<<G32, 32, 0, stream>>>(M1, st, SigX, DD, DD, DD);
  gemm_nn32_kernel<<<G32, 32, 0, stream>>>(G, M1, st, DD, DD, DD);
  symmetrize_kernel<<<EW, 256, 0, stream>>>(Msym, G);
  sumsq_kernel<<<256, 256, 0, stream>>>(scal + 3, Msym, DD * DD);
  ns_init_kernel<<<EW, 256, 0, stream>>>(Yb, Zb, Msym, scal + 3);
  {
    float *Y = Yb, *Z = Zb, *Yn2 = Ynb, *Zn2 = Znb;
    for (int it = 0; it < NS_ITERS; ++it) {
      gemm_nn32_kernel<<<G32, 32, 0, stream>>>(Pb, Z, Y, DD, DD, DD);
      ns_tk_kernel<<<EW, 256, 0, stream>>>(Pb);
      gemm_nn32_kernel<<<G32, 32, 0, stream>>>(Yn2, Y, Pb, DD, DD, DD);
      gemm_nn32_kernel<<<G32, 32, 0, stream>>>(Zn2, Pb, Z, DD, DD, DD);
      float* t1 = Y; Y = Yn2; Yn2 = t1;
      float* t2 = Z; Z = Zn2; Zn2 = t2;
    }
    trace_kernel<<<1, 256, 0, stream>>>(scal + 4, Y);  // * ssq2^{1/4} in combine
  }

  // -------- mean term + W regularizer --------
  meandiff_kernel<<<1, 256, 0, stream>>>(scal + 5, muT, muX);
  sumsq_kernel<<<256, 256, 0, stream>>>(scal + 7, W, DD * DD);

  // -------- kNN: chunked distances + top-K merge --------
  const int NCHUNK = (NX + NCC - 1) / NCC;  // 25
  for (int ch = 0; ch < NCHUNK; ++ch) {
    int j0 = ch * NCC;
    int jcount = (NX - j0 < NCC) ? (NX - j0) : NCC;
    gemm_d2_kernel<<<(BQ / 32) * (NCC / 32), 32, 0, stream>>>(d2c, T, X, Tn, Xn,
                                                              j0, jcount);
    topk_merge_kernel<<<BQ, 256, 0, stream>>>(topkd, topki, d2c, j0);
  }

  // -------- post weights, cost matrix, median, Sinkhorn --------
  postw_kernel<<<BQ, 32, 0, stream>>>(postw, topkd);
  costmat_kernel<<<BQ, 256, 0, stream>>>(Craw, X, Xn, topki, pre_indices, q_indices);
  median_kernel<<<BQ, 256, 0, stream>>>(med, Craw);
  sinkhorn_kernel<<<BQ, 64, 0, stream>>>(scal + 6, Craw, med, postw, pre_weights,
                                         q_indices);

  // -------- combine --------
  combine_kernel<<<1, 32, 0, stream>>>(out, scal + 0, scal + 1, scal + 4, scal + 3,
                                       scal + 5, scal + 6, scal + 7);
}